// BatchTopK_17197049053506
// MI455X (gfx1250) — compile-verified
//
#include <hip/hip_runtime.h>

#ifndef __has_builtin
#define __has_builtin(x) 0
#endif

#define BK_NBINS_MAX 2048
#define BK_NREP 4
#define BK_THREADS 256

typedef const __attribute__((address_space(1))) void* bk_as1cv;
typedef __attribute__((address_space(3))) void* bk_as3v;
typedef int bk_v4i __attribute__((ext_vector_type(4)));
typedef __attribute__((address_space(1))) bk_v4i* bk_gp4;
typedef __attribute__((address_space(3))) bk_v4i* bk_lp4;

// ---- gfx1250 async global->LDS copy (ASYNCcnt path), with fallbacks ----
#if __has_builtin(__builtin_amdgcn_global_load_async_to_lds_b128)
#define BK_ASYNC 1
__device__ __forceinline__ void bk_async16(const float4* g, float4* l) {
  // signature (probe-confirmed by diagnostic): (v4i AS1*, v4i AS3*, i32, i32)
  __builtin_amdgcn_global_load_async_to_lds_b128(
      (bk_gp4)(bk_as1cv)g, (bk_lp4)(bk_as3v)l, 0, 0);
}
#elif defined(__gfx1250__)
#define BK_ASYNC 1
__device__ __forceinline__ void bk_async16(const float4* g, float4* l) {
  unsigned loff = (unsigned)(unsigned long long)(bk_as3v)l;
  unsigned long long ga = (unsigned long long)g;
  asm volatile("global_load_async_to_lds_b128 %0, %1, off"
               :: "v"(loff), "v"(ga) : "memory");
}
#else
#define BK_ASYNC 0
__device__ __forceinline__ void bk_async16(const float4* g, float4* l) { *l = *g; }
#endif

#if __has_builtin(__builtin_amdgcn_s_wait_asynccnt)
#define BK_WAITA(n) do { __builtin_amdgcn_s_wait_asynccnt(n); asm volatile("" ::: "memory"); } while (0)
#elif defined(__gfx1250__)
#define BK_WAITA(n) asm volatile("s_wait_asynccnt " #n ::: "memory")
#else
#define BK_WAITA(n) ((void)0)
#endif

// Order-preserving float -> uint key: larger key <=> larger float.
__device__ __forceinline__ unsigned bk_key(float f) {
  unsigned u = __float_as_uint(f);
  return u ^ ((unsigned)((int)u >> 31) | 0x80000000u);
}

// ws layout (uint32): [0..2047] histogram bins, [2048] prefix,
// [2049] k_remaining, [2050] threshold float bits.

__global__ __launch_bounds__(BK_THREADS) void bk_init(unsigned* __restrict__ ws, unsigned k) {
  for (int j = threadIdx.x; j < 2051; j += BK_THREADS)
    ws[j] = (j == 2049) ? k : 0u;
}

__global__ __launch_bounds__(BK_THREADS) void bk_hist(
    const float* __restrict__ x, unsigned* __restrict__ ws,
    int numTiles, int shift, int bits, int pass, long long n)
{
  __shared__ unsigned lh[BK_NBINS_MAX * BK_NREP];
#if BK_ASYNC
  __shared__ float4 stage[2][BK_THREADS];
#endif
  const int tid = threadIdx.x;
  const unsigned prefix = ws[2048];
  const int prefshift = shift + bits;
  const unsigned mask = (1u << bits) - 1u;
  const int rep = tid & (BK_NREP - 1);

  for (int i = tid; i < BK_NBINS_MAX * BK_NREP; i += BK_THREADS) lh[i] = 0u;
  __syncthreads();

  const float4* x4 = (const float4*)x;

#define BK_PROC(fv) do {                                                   \
    unsigned _k = bk_key(fv);                                              \
    if (pass == 0 || (_k >> prefshift) == prefix)                          \
      atomicAdd(&lh[((_k >> shift) & mask) * BK_NREP + rep], 1u);          \
  } while (0)

  const int t0 = (int)blockIdx.x;
#if BK_ASYNC
  int cur = 0;
  if (t0 < numTiles)
    bk_async16(x4 + (size_t)t0 * BK_THREADS + tid, &stage[0][tid]);
  for (int t = t0; t < numTiles; t += (int)gridDim.x) {
    int tn = t + (int)gridDim.x;
    if (tn < numTiles) {
      bk_async16(x4 + (size_t)tn * BK_THREADS + tid, &stage[cur ^ 1][tid]);
      BK_WAITA(1);
    } else {
      BK_WAITA(0);
    }
    asm volatile("" ::: "memory");
    float4 v = stage[cur][tid];
    cur ^= 1;
    BK_PROC(v.x); BK_PROC(v.y); BK_PROC(v.z); BK_PROC(v.w);
  }
#else
  for (int t = t0; t < numTiles; t += (int)gridDim.x) {
    float4 v = x4[(size_t)t * BK_THREADS + tid];
    BK_PROC(v.x); BK_PROC(v.y); BK_PROC(v.z); BK_PROC(v.w);
  }
#endif
  // scalar tail (n not a multiple of 1024 per tile)
  for (long long i = (long long)numTiles * (BK_THREADS * 4) +
                     (long long)blockIdx.x * BK_THREADS + tid;
       i < n; i += (long long)gridDim.x * BK_THREADS) {
    float f = x[i];
    BK_PROC(f);
  }
#undef BK_PROC

  __syncthreads();
  const int nbins = 1 << bits;
  for (int b = tid; b < nbins; b += BK_THREADS) {
    unsigned s = lh[b * BK_NREP + 0] + lh[b * BK_NREP + 1] +
                 lh[b * BK_NREP + 2] + lh[b * BK_NREP + 3];
    if (s) atomicAdd(&ws[b], s);
  }
}

__global__ __launch_bounds__(BK_THREADS) void bk_select(
    unsigned* __restrict__ ws, int bits, int finalPass)
{
  __shared__ unsigned sh[BK_NBINS_MAX];
  __shared__ unsigned part[64];
  const int tid = threadIdx.x;
  const int nbins = 1 << bits;

  for (int i = tid; i < BK_NBINS_MAX; i += BK_THREADS)
    sh[i] = (i < nbins) ? ws[i] : 0u;
  __syncthreads();
  // zero global histogram for the next pass
  for (int i = tid; i < BK_NBINS_MAX; i += BK_THREADS) ws[i] = 0u;

  const int ngroups = nbins >> 5;
  if (tid < ngroups) {
    unsigned s = 0;
    for (int j = 0; j < 32; ++j) s += sh[tid * 32 + j];
    part[tid] = s;
  }
  __syncthreads();

  if (tid == 0) {
    unsigned krem = ws[2049];
    unsigned prefix = ws[2048];
    unsigned acc = 0;
    int g = ngroups - 1;
    for (; g > 0; --g) {
      if (acc + part[g] >= krem) break;
      acc += part[g];
    }
    int d = g * 32;
    for (int j = 31; j >= 0; --j) {
      unsigned c = sh[g * 32 + j];
      if (acc + c >= krem) { d = g * 32 + j; break; }
      acc += c;
    }
    krem -= acc;                       // rank within selected digit
    prefix = (prefix << bits) | (unsigned)d;
    if (finalPass) {
      unsigned key = prefix;           // full 32-bit key of k-th largest
      unsigned u = (key & 0x80000000u) ? (key ^ 0x80000000u) : ~key;
      ws[2050] = u;                    // threshold as float bits
    } else {
      ws[2048] = prefix;
      ws[2049] = krem;
    }
  }
}

__global__ __launch_bounds__(BK_THREADS) void bk_apply(
    const float* __restrict__ x, float* __restrict__ out,
    const unsigned* __restrict__ ws, long long n)
{
  const float thr = __uint_as_float(ws[2050]);
  const long long n4 = n >> 2;
  const float4* x4 = (const float4*)x;
  float4* o4 = (float4*)out;
  const long long stride = (long long)gridDim.x * blockDim.x;
  for (long long i = (long long)blockIdx.x * blockDim.x + threadIdx.x;
       i < n4; i += stride) {
    __builtin_prefetch(&x4[i + stride], 0, 0);
    float4 v = x4[i];
    float4 r;
    r.x = (v.x >= thr) ? fmaxf(v.x, 0.0f) : 0.0f;
    r.y = (v.y >= thr) ? fmaxf(v.y, 0.0f) : 0.0f;
    r.z = (v.z >= thr) ? fmaxf(v.z, 0.0f) : 0.0f;
    r.w = (v.w >= thr) ? fmaxf(v.w, 0.0f) : 0.0f;
    o4[i] = r;
  }
  for (long long i = n4 * 4 + (long long)blockIdx.x * blockDim.x + threadIdx.x;
       i < n; i += stride) {
    float v = x[i];
    out[i] = (v >= thr) ? fmaxf(v, 0.0f) : 0.0f;
  }
}

extern "C" void kernel_launch(void* const* d_in, const int* in_sizes, int n_in,
                              void* d_out, int out_size, void* d_ws, size_t ws_size,
                              hipStream_t stream) {
  (void)n_in; (void)out_size; (void)ws_size;
  const float* x = (const float*)d_in[0];
  float* out = (float*)d_out;
  unsigned* ws = (unsigned*)d_ws;

  const long long n = (long long)in_sizes[0];      // 4096 * 24576
  long long rows = n / 24576; if (rows < 1) rows = 1;
  long long tk = 64LL * rows; if (tk > n) tk = n;  // total_k = min(K*batch, size)
  const unsigned total_k = (unsigned)tk;
  const int numTiles = (int)(n / (BK_THREADS * 4)); // 1024 floats per tile

  const int HGRID = 4096;
  const int AGRID = 8192;

  bk_init<<<1, BK_THREADS, 0, stream>>>(ws, total_k);
  // pass 0: key bits [31:21] (11 bits)
  bk_hist<<<HGRID, BK_THREADS, 0, stream>>>(x, ws, numTiles, 21, 11, 0, n);
  bk_select<<<1, BK_THREADS, 0, stream>>>(ws, 11, 0);
  // pass 1: key bits [20:10] (11 bits)
  bk_hist<<<HGRID, BK_THREADS, 0, stream>>>(x, ws, numTiles, 10, 11, 1, n);
  bk_select<<<1, BK_THREADS, 0, stream>>>(ws, 11, 0);
  // pass 2: key bits [9:0] (10 bits)
  bk_hist<<<HGRID, BK_THREADS, 0, stream>>>(x, ws, numTiles, 0, 10, 2, n);
  bk_select<<<1, BK_THREADS, 0, stream>>>(ws, 10, 1);
  // elementwise gate
  bk_apply<<<AGRID, BK_THREADS, 0, stream>>>(x, out, ws, n);
}